// ContrastiveResNetGCN_59725815218593
// MI455X (gfx1250) — compile-verified
//
#include <hip/hip_runtime.h>
#include <hip/hip_bf16.h>
#include <stdint.h>

typedef float v2f __attribute__((ext_vector_type(2)));
typedef float v8f __attribute__((ext_vector_type(8)));

// D = A(16x4,f32) * B(4x16,f32) + C(16x16,f32)
#define WMMA_F32(a, b, c)                                                     \
  __builtin_amdgcn_wmma_f32_16x16x4_f32(false, (a), false, (b), (short)0,     \
                                        (c), false, false)

#if __has_builtin(__builtin_amdgcn_global_load_async_to_lds_b128)
#define USE_ASYNC_COPY 1
#else
#define USE_ASYNC_COPY 0
#endif

#if USE_ASYNC_COPY
// Builtin signature (from hipcc diagnostic): param 0 is
// '__attribute__((vector_size(16))) int *' in the global address space,
// param 1 the LDS-side counterpart, then two immediate ints (offset, cpol).
typedef int v4i_vs __attribute__((vector_size(16)));
typedef __attribute__((address_space(1))) v4i_vs as1_v4i;
typedef __attribute__((address_space(3))) v4i_vs as3_v4i;

// generic->global: same 64-bit address, reinterpret through uintptr_t.
__device__ __forceinline__ as1_v4i* to_gbl(const void* p) {
  return (as1_v4i*)(uintptr_t)p;
}
// generic->LDS: low 32 bits of a generic shared-memory address are the LDS
// offset (flat aperture truncation); integer->AS3 pointer cast is a plain
// reinterpret, yielding a valid 32-bit local pointer.
__device__ __forceinline__ as3_v4i* to_lds(void* p) {
  return (as3_v4i*)(uint32_t)(uintptr_t)p;
}
__device__ __forceinline__ void wait_async0() {
#if __has_builtin(__builtin_amdgcn_s_wait_asynccnt)
  __builtin_amdgcn_s_wait_asynccnt(0);
#else
  asm volatile("s_wait_asynccnt 0x0" ::: "memory");
#endif
}
#endif

// ---------------------------------------------------------------------------
// LDS double-buffered GEMM: C[M,N] = epilogue(A[M,K] @ B[K,N]), row-major.
// Block = 256 threads (8 waves) -> 128x64 C-tile, BK=16 k-chunks.
// sA padded to stride 20 (conflict-free ds_load_b64 A-fragments, 16B-aligned
// b128 stores), sB stride 68. Async global->LDS copies (ASYNCcnt) when
// available; software-pipelined synchronous staging otherwise.
// MODE: 0 = none, 1 = relu, 2 = out = alpha*rowScale[row]*acc + bias[col]
// ---------------------------------------------------------------------------
template <int MODE>
__global__ __launch_bounds__(256) void gemm128x64(
    const float* __restrict__ A, const float* __restrict__ B,
    float* __restrict__ C, int M, int K, int N,
    const float* __restrict__ rowScale, const float* __restrict__ bias,
    float alpha) {
  __shared__ float sA[2][128 * 20];  // [buf][row*20 + k], 20 KB
  __shared__ float sB[2][16 * 68];   // [buf][k*68 + col], 8.7 KB

  const int t    = threadIdx.x;
  const int lane = t & 31;
  const int wave = t >> 5;
  const int m    = lane & 15;         // row (A) / col (B,C) within tile
  const int kh   = (lane >> 4) << 1;  // 0 or 2
  const int rowBase = blockIdx.y * 128;
  const int col0    = blockIdx.x * 64;

  // staging assignment (coalesced b128 per thread)
  const int ar  = t >> 2;         // 0..63   (A rows ar and ar+64)
  const int ac4 = (t & 3) * 4;    // 0,4,8,12
  const int bk  = t >> 4;         // 0..15
  const int bc4 = (t & 15) * 4;   // 0..60

  const float* gA0 = A + (size_t)(rowBase + ar) * K + ac4;
  const float* gA1 = gA0 + (size_t)64 * K;
  const float* gB  = B + (size_t)bk * N + col0 + bc4;

  const int sAo0 = ar * 20 + ac4;
  const int sAo1 = (ar + 64) * 20 + ac4;
  const int sBo  = bk * 68 + bc4;

  const int nch = K / 16;

  v8f acc0 = {}, acc1 = {}, acc2 = {}, acc3 = {};

#if USE_ASYNC_COPY
  __builtin_amdgcn_global_load_async_to_lds_b128(to_gbl(gA0),
                                                 to_lds(&sA[0][sAo0]), 0, 0);
  __builtin_amdgcn_global_load_async_to_lds_b128(to_gbl(gA1),
                                                 to_lds(&sA[0][sAo1]), 0, 0);
  __builtin_amdgcn_global_load_async_to_lds_b128(to_gbl(gB),
                                                 to_lds(&sB[0][sBo]), 0, 0);
#else
  float4 ra0 = *(const float4*)gA0;
  float4 ra1 = *(const float4*)gA1;
  float4 rb  = *(const float4*)gB;
  *(float4*)&sA[0][sAo0] = ra0;
  *(float4*)&sA[0][sAo1] = ra1;
  *(float4*)&sB[0][sBo]  = rb;
  __syncthreads();
#endif

  for (int ch = 0; ch < nch; ++ch) {
    const int buf = ch & 1;
    const bool more = (ch + 1 < nch);

#if USE_ASYNC_COPY
    wait_async0();       // this wave's copies for chunk `ch` have landed
    __syncthreads();     // ... so everyone's have; prior reads of buf^1 done
    if (more) {
      const size_t ka = (size_t)(ch + 1) * 16;
      const int nb = buf ^ 1;
      __builtin_amdgcn_global_load_async_to_lds_b128(
          to_gbl(gA0 + ka), to_lds(&sA[nb][sAo0]), 0, 0);
      __builtin_amdgcn_global_load_async_to_lds_b128(
          to_gbl(gA1 + ka), to_lds(&sA[nb][sAo1]), 0, 0);
      __builtin_amdgcn_global_load_async_to_lds_b128(
          to_gbl(gB + ka * N), to_lds(&sB[nb][sBo]), 0, 0);
    }
#else
    if (more) {  // prefetch next chunk into registers during compute
      const size_t ka = (size_t)(ch + 1) * 16;
      ra0 = *(const float4*)(gA0 + ka);
      ra1 = *(const float4*)(gA1 + ka);
      rb  = *(const float4*)(gB + ka * N);
    }
#endif

    // compute chunk `ch`: 4 k-steps x 4 column tiles
    const float* pA = &sA[buf][(wave * 16 + m) * 20 + kh];
    const float* pB = &sB[buf][kh * 68 + m];
#pragma unroll
    for (int ks = 0; ks < 4; ++ks) {
      v2f a = *(const v2f*)(pA + ks * 4);
      const float* br = pB + ks * 4 * 68;
      v2f b0, b1, b2, b3;
      b0.x = br[0];      b0.y = br[68];
      b1.x = br[16];     b1.y = br[68 + 16];
      b2.x = br[32];     b2.y = br[68 + 32];
      b3.x = br[48];     b3.y = br[68 + 48];
      acc0 = WMMA_F32(a, b0, acc0);
      acc1 = WMMA_F32(a, b1, acc1);
      acc2 = WMMA_F32(a, b2, acc2);
      acc3 = WMMA_F32(a, b3, acc3);
    }

#if !USE_ASYNC_COPY
    __syncthreads();
    if (more) {
      *(float4*)&sA[buf ^ 1][sAo0] = ra0;
      *(float4*)&sA[buf ^ 1][sAo1] = ra1;
      *(float4*)&sB[buf ^ 1][sBo]  = rb;
      __syncthreads();
    }
#endif
  }

  float bb0 = 0.f, bb1 = 0.f, bb2 = 0.f, bb3 = 0.f;
  if (MODE == 2) {
    bb0 = bias[col0 + m];
    bb1 = bias[col0 + m + 16];
    bb2 = bias[col0 + m + 32];
    bb3 = bias[col0 + m + 48];
  }

#pragma unroll
  for (int v = 0; v < 8; ++v) {
    const int row = rowBase + wave * 16 + v + 8 * (lane >> 4);
    float x0 = acc0[v], x1 = acc1[v], x2 = acc2[v], x3 = acc3[v];
    if (MODE == 1) {
      x0 = fmaxf(x0, 0.f); x1 = fmaxf(x1, 0.f);
      x2 = fmaxf(x2, 0.f); x3 = fmaxf(x3, 0.f);
    } else if (MODE == 2) {
      const float s = alpha * rowScale[row];
      x0 = x0 * s + bb0; x1 = x1 * s + bb1;
      x2 = x2 * s + bb2; x3 = x3 * s + bb3;
    }
    float* crow = C + (size_t)row * N + col0 + m;
    crow[0] = x0; crow[16] = x1; crow[32] = x2; crow[48] = x3;
  }
}

// ---------------------------------------------------------------------------
// r[n] = 1/||Xp[n,:]|| (P=128), Z[n,:] = r[n]*h[n,:] (H=128). Wave per row.
// ---------------------------------------------------------------------------
__global__ __launch_bounds__(256) void rownorm_scale(
    const float* __restrict__ Xp, const float* __restrict__ h,
    float* __restrict__ r, float* __restrict__ Z) {
  const int lane = threadIdx.x & 31;
  const int wave = threadIdx.x >> 5;
  const int row  = blockIdx.x * 8 + wave;

  const float* xr = Xp + (size_t)row * 128;
  float s = 0.f;
#pragma unroll
  for (int i = 0; i < 4; ++i) {
    const float v = xr[lane + 32 * i];
    s += v * v;
  }
#pragma unroll
  for (int off = 16; off; off >>= 1) s += __shfl_xor(s, off, 32);

  const float rv = rsqrtf(fmaxf(s, 1e-16f));
  if (lane == 0) r[row] = rv;

  const float* hr = h + (size_t)row * 128;
  float* zr = Z + (size_t)row * 128;
#pragma unroll
  for (int i = 0; i < 4; ++i) zr[lane + 32 * i] = rv * hr[lane + 32 * i];
}

// ---------------------------------------------------------------------------
// T[128,128] += Xp^T @ Z, split-K, float atomics; transposed-A access is
// lane-contiguous since Xp is row-major [K,128]. One wave per (tile, chunk).
// ---------------------------------------------------------------------------
__global__ __launch_bounds__(32) void atb_splitk(
    const float* __restrict__ Xp, const float* __restrict__ Z,
    float* __restrict__ T, int kChunk) {
  const int lane = threadIdx.x & 31;
  const int m    = lane & 15;
  const int kh   = (lane >> 4) << 1;
  const int p0   = blockIdx.x * 16;
  const int q0   = blockIdx.y * 16;
  const int kBeg = blockIdx.z * kChunk;

  v8f acc = {};
  for (int k0 = kBeg; k0 < kBeg + kChunk; k0 += 4) {
    const float* xk = Xp + (size_t)(k0 + kh) * 128;
    const float* zk = Z + (size_t)(k0 + kh) * 128;
    v2f a, b;
    a.x = xk[p0 + m];
    a.y = xk[128 + p0 + m];
    b.x = zk[q0 + m];
    b.y = zk[128 + q0 + m];
    acc = WMMA_F32(a, b, acc);
  }
#pragma unroll
  for (int v = 0; v < 8; ++v) {
    const int p = p0 + v + 8 * (lane >> 4);
    atomicAdd(&T[p * 128 + q0 + m], acc[v]);
  }
}

__global__ __launch_bounds__(256) void zero_kernel(float* __restrict__ p,
                                                   int n) {
  const int i = blockIdx.x * 256 + threadIdx.x;
  if (i < n) p[i] = 0.f;
}

// ---------------------------------------------------------------------------
// out = (1/N) * D^{-1} Xp (Xp^T (D^{-1} (x1@gc_w))) + gc_b,
// Xp = relu(x1@W1)@W2, D = diag(||xp_i||)  — equal to the reference's
// cosine-graph conv (clamp inactive) without the 8192x8192 matrix.
// ---------------------------------------------------------------------------
extern "C" void kernel_launch(void* const* d_in, const int* in_sizes, int n_in,
                              void* d_out, int out_size, void* d_ws,
                              size_t ws_size, hipStream_t stream) {
  (void)in_sizes; (void)n_in; (void)out_size; (void)ws_size;
  const int N = 8192, D = 512, P = 128, H = 128;

  const float* x1  = (const float*)d_in[0];  // [N,D]
  const float* w1  = (const float*)d_in[1];  // [D,D]
  const float* w2  = (const float*)d_in[2];  // [D,P]
  const float* gcw = (const float*)d_in[3];  // [D,H]
  const float* gcb = (const float*)d_in[4];  // [H]
  float* out = (float*)d_out;                // [N,H]

  float* H1 = (float*)d_ws;                  // [N,D] 16 MB
  float* Xp = H1 + (size_t)N * D;            // [N,P]  4 MB
  float* hB = Xp + (size_t)N * P;            // [N,H]  4 MB
  float* Zb = hB + (size_t)N * H;            // [N,H]  4 MB
  float* rb = Zb + (size_t)N * H;            // [N]
  float* Tb = rb + N;                        // [P,H]

  dim3 blk(256);

  // 1) H1 = relu(x1 @ W1)         M=8192 K=512 N=512
  gemm128x64<1><<<dim3(D / 64, N / 128), blk, 0, stream>>>(
      x1, w1, H1, N, D, D, nullptr, nullptr, 0.f);
  // 2) Xp = H1 @ W2               M=8192 K=512 N=128
  gemm128x64<0><<<dim3(P / 64, N / 128), blk, 0, stream>>>(
      H1, w2, Xp, N, D, P, nullptr, nullptr, 0.f);
  // 3) h = x1 @ gc_w              M=8192 K=512 N=128
  gemm128x64<0><<<dim3(H / 64, N / 128), blk, 0, stream>>>(
      x1, gcw, hB, N, D, H, nullptr, nullptr, 0.f);
  // 4) r = 1/||Xp row||, Z = r .* h
  rownorm_scale<<<dim3(N / 8), blk, 0, stream>>>(Xp, hB, rb, Zb);
  // 5) T = 0
  zero_kernel<<<dim3((P * H + 255) / 256), blk, 0, stream>>>(Tb, P * H);
  // 6) T += Xp^T @ Z              split-K = 32 (chunk 256)
  atb_splitk<<<dim3(P / 16, H / 16, 32), dim3(32), 0, stream>>>(Xp, Zb, Tb,
                                                                N / 32);
  // 7) out = (1/N) * r .* (Xp @ T) + gc_b
  gemm128x64<2><<<dim3(H / 64, N / 128), blk, 0, stream>>>(
      Xp, Tb, out, N, P, H, rb, gcb, 1.0f / (float)N);
}